// MessagePassingLayer_33956011442595
// MI455X (gfx1250) — compile-verified
//
#include <hip/hip_runtime.h>
#include <math.h>

#define NC 32
#define NRBF 16

typedef __attribute__((ext_vector_type(2))) float v2f;
typedef __attribute__((ext_vector_type(8))) float v8f;

__device__ __forceinline__ float silu_f(float x) { return x / (1.0f + __expf(-x)); }

// 13 floats per channel: [0]=l0, [1..3]=l1, [4..12]=l2 (row-major 3x3)
struct V13 { float v[13]; };

// LDS layout for one node's 32-channel feature set: s0[0..31], s1[32..127] (c*3+u), s2[128..415] (c*9+p)
__device__ __forceinline__ V13 lmix_lane(const float* __restrict__ W, const float* __restrict__ s, int f) {
  V13 y;
#pragma unroll
  for (int i = 0; i < 13; ++i) y.v[i] = 0.f;
  const float* W0 = W + f * NC;
  const float* W1 = W + NC * NC + f * NC;
  const float* W2 = W + 2 * NC * NC + f * NC;
  for (int c = 0; c < NC; ++c) {
    float w0 = W0[c], w1 = W1[c], w2 = W2[c];
    y.v[0] += w0 * s[c];
#pragma unroll
    for (int u = 0; u < 3; ++u) y.v[1 + u] += w1 * s[NC + c * 3 + u];
#pragma unroll
    for (int p = 0; p < 9; ++p) y.v[4 + p] += w2 * s[NC + 3 * NC + c * 9 + p];
  }
  return y;
}

// Premix table layout: P[node][m][comp][f], m in 0..5 <-> lin_mix_w[2..7]
// Per-lane read: 13 coalesced b32 loads (stride 32 floats).
__device__ __forceinline__ V13 load_premix(const float* __restrict__ P, int node, int m, int f) {
  const float* p = P + (size_t)node * 2496 + m * 416 + f;
  V13 y;
#pragma unroll
  for (int c = 0; c < 13; ++c) y.v[c] = p[c * 32];
  return y;
}

// Full 11-combo tensor product (mx=my=mz=2), weights tw[ci*NC + f]
__device__ __forceinline__ V13 tp_lane(const V13& x, const V13& y, const float* __restrict__ tw, int f) {
  float w[11];
#pragma unroll
  for (int i = 0; i < 11; ++i) w[i] = tw[i * NC + f];
  const float x0 = x.v[0]; const float* x1 = x.v + 1; const float* x2 = x.v + 4;
  const float y0 = y.v[0]; const float* y1 = y.v + 1; const float* y2 = y.v + 4;
  V13 o;
  float d11 = x1[0] * y1[0] + x1[1] * y1[1] + x1[2] * y1[2];
  float d22 = 0.f;
#pragma unroll
  for (int p = 0; p < 9; ++p) d22 += x2[p] * y2[p];
  o.v[0] = w[0] * x0 * y0 + w[5] * d11 + w[10] * d22;
#pragma unroll
  for (int u = 0; u < 3; ++u) {
    float s6 = x1[0] * y2[u * 3 + 0] + x1[1] * y2[u * 3 + 1] + x1[2] * y2[u * 3 + 2]; // 'ecu,ecvu->ecv'
    float s8 = x2[u * 3 + 0] * y1[0] + x2[u * 3 + 1] * y1[1] + x2[u * 3 + 2] * y1[2]; // 'ecuv,ecv->ecu'
    o.v[1 + u] = w[1] * x0 * y1[u] + w[3] * x1[u] * y0 + w[6] * s6 + w[8] * s8;
  }
#pragma unroll
  for (int u = 0; u < 3; ++u)
#pragma unroll
    for (int vv = 0; vv < 3; ++vv) {
      float s9 = x2[u * 3 + 0] * y2[vv * 3 + 0] + x2[u * 3 + 1] * y2[vv * 3 + 1] + x2[u * 3 + 2] * y2[vv * 3 + 2]; // 'ecuv,ecwv->ecuw'
      o.v[4 + u * 3 + vv] = w[2] * x0 * y2[u * 3 + vv] + w[4] * x1[u] * y1[vv] + w[7] * x2[u * 3 + vv] * y0 + w[9] * s9;
    }
  return o;
}

// ---------------- K0: RMS norm per node, per l ----------------
__global__ void norm_kernel(const float* __restrict__ nf0, const float* __restrict__ nf1,
                            const float* __restrict__ nf2, float* __restrict__ o0,
                            float* __restrict__ o1, float* __restrict__ o2) {
  __shared__ float red[128];
  const int n = blockIdx.x, t = threadIdx.x;
  float sums[3] = {0.f, 0.f, 0.f};
  for (int i = t; i < 32; i += 128) { float v = nf0[n * 32 + i]; sums[0] += v * v; }
  for (int i = t; i < 96; i += 128) { float v = nf1[n * 96 + i]; sums[1] += v * v; }
  for (int i = t; i < 288; i += 128) { float v = nf2[n * 288 + i]; sums[2] += v * v; }
  float scale[3];
  const float cnt[3] = {32.f, 96.f, 288.f};
#pragma unroll
  for (int l = 0; l < 3; ++l) {
    red[t] = sums[l];
    __syncthreads();
    for (int s = 64; s > 0; s >>= 1) { if (t < s) red[t] += red[t + s]; __syncthreads(); }
    float rms = sqrtf(red[0] / cnt[l]);
    scale[l] = 1.0f / (1.0f + rms);
    __syncthreads();
  }
  for (int i = t; i < 32; i += 128) o0[n * 32 + i] = nf0[n * 32 + i] * scale[0];
  for (int i = t; i < 96; i += 128) o1[n * 96 + i] = nf1[n * 96 + i] * scale[1];
  for (int i = t; i < 288; i += 128) o2[n * 288 + i] = nf2[n * 288 + i] * scale[2];
}

// comb layout per node (2080 floats): l0 at [c], l1 at [160 + c*3+u], l2 at [640 + c*9+p], c in [0,160)
// channels 0..31 = mii TP, 32..63 = nf, 64..159 = msg (scaled by 0.1 at readout)

// ---------------- K1: per-node premix of lin_mix_w[2..7] (hoisted out of edges) ----------------
__global__ void node_premix_kernel(const float* __restrict__ nrm0, const float* __restrict__ nrm1,
                                   const float* __restrict__ nrm2, const float* __restrict__ lin,
                                   float* __restrict__ P) {
  __shared__ float sN[416];
  const int n = blockIdx.x, f = threadIdx.x;
  sN[f] = nrm0[n * NC + f];
#pragma unroll
  for (int u = 0; u < 3; ++u) sN[32 + f * 3 + u] = nrm1[(n * NC + f) * 3 + u];
#pragma unroll
  for (int p = 0; p < 9; ++p) sN[128 + f * 9 + p] = nrm2[(n * NC + f) * 9 + p];
  __syncthreads();
  float* base = P + (size_t)n * 2496 + f;
  for (int m = 2; m <= 7; ++m) {
    V13 y = lmix_lane(lin + m * 3072, sN, f);
    float* p = base + (m - 2) * 416;
#pragma unroll
    for (int c = 0; c < 13; ++c) p[c * 32] = y.v[c];
  }
}

// ---------------- K2: node self-interaction (mii) ----------------
__global__ void node_mii_kernel(const float* __restrict__ nrm0, const float* __restrict__ nrm1,
                                const float* __restrict__ nrm2, const float* __restrict__ lin,
                                const float* __restrict__ tpw, float* __restrict__ comb) {
  __shared__ float sN[416];
  const int n = blockIdx.x, f = threadIdx.x;
  sN[f] = nrm0[n * NC + f];
#pragma unroll
  for (int u = 0; u < 3; ++u) sN[32 + f * 3 + u] = nrm1[(n * NC + f) * 3 + u];
#pragma unroll
  for (int p = 0; p < 9; ++p) sN[128 + f * 9 + p] = nrm2[(n * NC + f) * 9 + p];
  __syncthreads();
  V13 x = lmix_lane(lin + 0 * 3072, sN, f);
  V13 y = lmix_lane(lin + 1 * 3072, sN, f);
  V13 o = tp_lane(x, y, tpw + 0 * 352, f);
  float* np = comb + (size_t)n * 2080;
  np[f] = o.v[0];
#pragma unroll
  for (int u = 0; u < 3; ++u) np[160 + f * 3 + u] = o.v[1 + u];
#pragma unroll
  for (int p = 0; p < 9; ++p) np[640 + f * 9 + p] = o.v[4 + p];
  np[32 + f] = sN[f];
#pragma unroll
  for (int u = 0; u < 3; ++u) np[160 + (32 + f) * 3 + u] = sN[32 + f * 3 + u];
#pragma unroll
  for (int p = 0; p < 9; ++p) np[640 + (32 + f) * 9 + p] = sN[128 + f * 9 + p];
}

// Shared per-edge pipeline: one wave per edge, channel-per-lane.
// All node-side lmixes come from the premix table; only lin_mix_w[8] (on em') is per-edge.
__device__ __forceinline__ void edge_pipeline(int e, int f, int ni, int nj,
    const float* __restrict__ P, const float* __restrict__ r_ij,
    const float* __restrict__ lin, const float* __restrict__ tpw,
    float* sT, V13& u, V13& tp1, V13& ta, V13& tb) {
  float rx = r_ij[e * 3 + 0], ry = r_ij[e * 3 + 1], rz = r_ij[e * 3 + 2];
  float inv = rsqrtf(rx * rx + ry * ry + rz * rz);
  float nh[3] = {rx * inv, ry * inv, rz * inv};
  u.v[0] = 1.0f;
#pragma unroll
  for (int i = 0; i < 3; ++i) u.v[1 + i] = nh[i];
#pragma unroll
  for (int i = 0; i < 3; ++i)
#pragma unroll
    for (int j = 0; j < 3; ++j) u.v[4 + i * 3 + j] = nh[i] * nh[j];

  V13 h2 = load_premix(P, nj, 3, f);                 // _lmix(hj, W5)
  V13 emp = tp_lane(u, h2, tpw + 2 * 352, f);        // _tp(u, h2, tp_w[2])
  sT[f] = emp.v[0];
#pragma unroll
  for (int uu = 0; uu < 3; ++uu) sT[32 + f * 3 + uu] = emp.v[1 + uu];
#pragma unroll
  for (int p = 0; p < 9; ++p) sT[128 + f * 9 + p] = emp.v[4 + p];
  __syncthreads();
  V13 em2 = lmix_lane(lin + 8 * 3072, sT, f);        // _lmix(em', W8)  (only per-edge lmix)
  V13 h1 = load_premix(P, ni, 2, f);                 // _lmix(hi, W4)
  tp1 = tp_lane(h1, em2, tpw + 1 * 352, f);          // _tp(h1, em, tp_w[1])
  V13 g6 = load_premix(P, ni, 4, f);                 // _lmix(hi, W6)
  ta = tp_lane(g6, u, tpw + 3 * 352, f);             // _tp(., u, tp_w[3])
  V13 g7 = load_premix(P, nj, 5, f);                 // _lmix(hj, W7)
  tb = tp_lane(u, g7, tpw + 4 * 352, f);             // _tp(u, ., tp_w[4])
}

// ---------------- K3: build att_feed (E x 192) ----------------
__global__ void edge_feed_kernel(const float* __restrict__ P, const float* __restrict__ r_ij,
                                 const int* __restrict__ edge_index, const float* __restrict__ lin,
                                 const float* __restrict__ tpw, const float* __restrict__ attw,
                                 float* __restrict__ feed, int E) {
  __shared__ float sT[416];
  const int e = blockIdx.x, f = threadIdx.x;
  const int ni = edge_index[e], nj = edge_index[E + e];
  V13 u, tp1, ta, tb;
  edge_pipeline(e, f, ni, nj, P, r_ij, lin, tpw, sT, u, tp1, ta, tb);
  // att0 = stacked combos (0,0,0,0),(1,1,1,0),(2,2,2,0) of tp(lmix(hi,W2), lmix(hj,W3))
  V13 ax = load_premix(P, ni, 0, f);
  V13 ay = load_premix(P, nj, 1, f);
  float d11 = ax.v[1] * ay.v[1] + ax.v[2] * ay.v[2] + ax.v[3] * ay.v[3];
  float d22 = 0.f;
#pragma unroll
  for (int p = 0; p < 9; ++p) d22 += ax.v[4 + p] * ay.v[4 + p];
  float* fd = feed + (size_t)e * 192;
  fd[f]        = attw[f] * ax.v[0] * ay.v[0];
  fd[32 + f]   = attw[32 + f] * d11;
  fd[64 + f]   = attw[64 + f] * d22;
  fd[96 + f]   = tp1.v[0];
  fd[128 + f]  = ta.v[0];
  fd[160 + f]  = tb.v[0];
}

// ---------------- K4: fused attention MLP + rbf via f32 WMMA ----------------
__device__ __forceinline__ v8f wmma4(v2f a, v2f b, v8f c) {
  return __builtin_amdgcn_wmma_f32_16x16x4_f32(false, a, false, b, (short)0, c, false, false);
}

__global__ void attn_wmma_kernel(const float* __restrict__ feed, const float* __restrict__ rbf,
                                 const float* __restrict__ w1, const float* __restrict__ b1,
                                 const float* __restrict__ w2, const float* __restrict__ b2,
                                 const float* __restrict__ w3, const float* __restrict__ b3,
                                 const float* __restrict__ rbfw, float* __restrict__ mult, int E) {
  __shared__ float sFeed[16 * 192];
  __shared__ float sH[16 * 96];
  __shared__ float sG[16 * 96];
  __shared__ float sR[16 * 16];
  const int tile = blockIdx.x;
  const int l = blockIdx.y;
  const int lane = threadIdx.x;
  const int e0 = tile * 16;
  for (int i = lane; i < 16 * 192; i += 32) sFeed[i] = feed[(size_t)e0 * 192 + i];
  for (int i = lane; i < 16 * 16; i += 32) sR[i] = rbf[(size_t)e0 * 16 + i];
  __syncthreads();
  const int row = lane & 15;          // A: edge row  |  B: output column
  const int kh = (lane >> 4) << 1;    // lanes 16-31 carry K=2,3 of each K-quad
  const int mBase = (lane >> 4) << 3; // C/D: M = r + 8*(lane>=16)
  const float* W1 = w1 + (size_t)l * 96 * 192;
  const float* W2 = w2 + (size_t)l * 96 * 96;
  const float* W3 = w3 + (size_t)l * 96 * 96;
  const float* RW = rbfw + (size_t)l * 96 * 16;

  // ---- layer 1: (16x192)*(192x96) ----
  v8f acc[6];
#pragma unroll
  for (int nt = 0; nt < 6; ++nt) {
    v8f c = {};
    const int nb = nt * 16;
    for (int kb = 0; kb < 192; kb += 4) {
      const int kk = kb + kh;
      v2f a; a[0] = sFeed[row * 192 + kk]; a[1] = sFeed[row * 192 + kk + 1];
      v2f b; b[0] = W1[(nb + row) * 192 + kk]; b[1] = W1[(nb + row) * 192 + kk + 1];
      c = wmma4(a, b, c);
    }
    acc[nt] = c;
  }
#pragma unroll
  for (int nt = 0; nt < 6; ++nt) {
    const int nb = nt * 16;
#pragma unroll
    for (int r = 0; r < 8; ++r) {
      int m = mBase + r, o = nb + row;
      sH[m * 96 + o] = silu_f(acc[nt][r] + b1[l * 96 + o]);
    }
  }
  __syncthreads();
  // ---- layer 2: (16x96)*(96x96) ----
#pragma unroll
  for (int nt = 0; nt < 6; ++nt) {
    v8f c = {};
    const int nb = nt * 16;
    for (int kb = 0; kb < 96; kb += 4) {
      const int kk = kb + kh;
      v2f a; a[0] = sH[row * 96 + kk]; a[1] = sH[row * 96 + kk + 1];
      v2f b; b[0] = W2[(nb + row) * 96 + kk]; b[1] = W2[(nb + row) * 96 + kk + 1];
      c = wmma4(a, b, c);
    }
    acc[nt] = c;
  }
  __syncthreads();
#pragma unroll
  for (int nt = 0; nt < 6; ++nt) {
    const int nb = nt * 16;
#pragma unroll
    for (int r = 0; r < 8; ++r) {
      int m = mBase + r, o = nb + row;
      sG[m * 96 + o] = silu_f(acc[nt][r] + b2[l * 96 + o]);
    }
  }
  __syncthreads();
  // ---- layer 3 + rbf mix, write a*rbfm ----
#pragma unroll
  for (int nt = 0; nt < 6; ++nt) {
    const int nb = nt * 16;
    v8f c3 = {};
    for (int kb = 0; kb < 96; kb += 4) {
      const int kk = kb + kh;
      v2f a; a[0] = sG[row * 96 + kk]; a[1] = sG[row * 96 + kk + 1];
      v2f b; b[0] = W3[(nb + row) * 96 + kk]; b[1] = W3[(nb + row) * 96 + kk + 1];
      c3 = wmma4(a, b, c3);
    }
    v8f cr = {};
    for (int kb = 0; kb < 16; kb += 4) {
      const int kk = kb + kh;
      v2f a; a[0] = sR[row * 16 + kk]; a[1] = sR[row * 16 + kk + 1];
      v2f b; b[0] = RW[(nb + row) * 16 + kk]; b[1] = RW[(nb + row) * 16 + kk + 1];
      cr = wmma4(a, b, cr);
    }
#pragma unroll
    for (int r = 0; r < 8; ++r) {
      int m = mBase + r, o = nb + row;
      float aval = c3[r] + b3[l * 96 + o];
      mult[((size_t)l * E + (e0 + m)) * 96 + o] = aval * cr[r];
    }
  }
}

// ---------------- K5: recompute em, scale by mult, scatter-add ----------------
__global__ void edge_msg_kernel(const float* __restrict__ P, const float* __restrict__ r_ij,
                                const int* __restrict__ edge_index, const float* __restrict__ lin,
                                const float* __restrict__ tpw, const float* __restrict__ mult,
                                float* __restrict__ comb, int E) {
  __shared__ float sT[416];
  const int e = blockIdx.x, f = threadIdx.x;
  const int ni = edge_index[e], nj = edge_index[E + e];
  V13 u, tp1, ta, tb;
  edge_pipeline(e, f, ni, nj, P, r_ij, lin, tpw, sT, u, tp1, ta, tb);
  float* nodep = comb + (size_t)ni * 2080;
#pragma unroll
  for (int g = 0; g < 3; ++g) {
    const V13& o = (g == 0) ? tp1 : ((g == 1) ? ta : tb);
    const int ch = 64 + g * 32 + f;   // msg channels 64..159
    const int oc = g * 32 + f;        // 0..95 within em
    float m0 = mult[((size_t)0 * E + e) * 96 + oc];
    float m1 = mult[((size_t)1 * E + e) * 96 + oc];
    float m2 = mult[((size_t)2 * E + e) * 96 + oc];
    atomicAdd(&nodep[ch], o.v[0] * m0);
#pragma unroll
    for (int uu = 0; uu < 3; ++uu) atomicAdd(&nodep[160 + ch * 3 + uu], o.v[1 + uu] * m1);
#pragma unroll
    for (int p = 0; p < 9; ++p) atomicAdd(&nodep[640 + ch * 9 + p], o.v[4 + p] * m2);
  }
}

// ---------------- K6: gate + final mix ----------------
__global__ void node_out_kernel(const float* __restrict__ comb, const float* __restrict__ mix_w,
                                const float* __restrict__ gate_w, const float* __restrict__ gate_b,
                                float* __restrict__ out) {
  __shared__ float sC[2080];
  const int n = blockIdx.x, f = threadIdx.x;
  const float* cp = comb + (size_t)n * 2080;
  for (int i = f; i < 2080; i += 32) {
    float v = cp[i];
    int c;
    if (i < 160) c = i;
    else if (i < 640) c = (i - 160) / 3;
    else c = (i - 640) / 9;
    sC[i] = (c >= 64) ? v * 0.1f : v;  // msg / AVG_NEIGHBORS
  }
  __syncthreads();
  float g = gate_b[f];
  for (int c = 0; c < 160; ++c) g += gate_w[f * 160 + c] * sC[c];
  g = silu_f(g);
  float a0 = 0.f;
  const float* M0 = mix_w + 0 * 32 * 160 + f * 160;
  for (int c = 0; c < 160; ++c) a0 += M0[c] * sC[c];
  out[(size_t)n * 416 + f] = a0 * g;
  float a1[3] = {0.f, 0.f, 0.f};
  const float* M1 = mix_w + 1 * 32 * 160 + f * 160;
  for (int c = 0; c < 160; ++c) {
    float w = M1[c];
#pragma unroll
    for (int uu = 0; uu < 3; ++uu) a1[uu] += w * sC[160 + c * 3 + uu];
  }
#pragma unroll
  for (int uu = 0; uu < 3; ++uu) out[(size_t)n * 416 + 32 + f * 3 + uu] = a1[uu] * g;
  float a2[9] = {0.f, 0.f, 0.f, 0.f, 0.f, 0.f, 0.f, 0.f, 0.f};
  const float* M2 = mix_w + 2 * 32 * 160 + f * 160;
  for (int c = 0; c < 160; ++c) {
    float w = M2[c];
#pragma unroll
    for (int p = 0; p < 9; ++p) a2[p] += w * sC[640 + c * 9 + p];
  }
#pragma unroll
  for (int p = 0; p < 9; ++p) out[(size_t)n * 416 + 128 + f * 9 + p] = a2[p] * g;
}

extern "C" void kernel_launch(void* const* d_in, const int* in_sizes, int n_in,
                              void* d_out, int out_size, void* d_ws, size_t ws_size,
                              hipStream_t stream) {
  (void)n_in; (void)out_size; (void)ws_size;
  const float* nf0     = (const float*)d_in[0];
  const float* nf1     = (const float*)d_in[1];
  const float* nf2     = (const float*)d_in[2];
  const float* r_ij    = (const float*)d_in[3];
  const float* rbf_ij  = (const float*)d_in[4];
  const int*   eidx    = (const int*)d_in[5];
  const float* lin     = (const float*)d_in[7];   // (9,3,32,32)
  const float* tpw     = (const float*)d_in[8];   // (5,11,32)
  const float* attw    = (const float*)d_in[9];   // (3,32)
  const float* rbfw    = (const float*)d_in[10];  // (3,96,16)
  const float* w1      = (const float*)d_in[11];  // (3,96,192)
  const float* b1      = (const float*)d_in[12];
  const float* w2      = (const float*)d_in[13];  // (3,96,96)
  const float* b2      = (const float*)d_in[14];
  const float* w3      = (const float*)d_in[15];  // (3,96,96)
  const float* b3      = (const float*)d_in[16];
  const float* mix_w   = (const float*)d_in[17];  // (3,32,160)
  const float* gate_w  = (const float*)d_in[18];  // (32,160)
  const float* gate_b  = (const float*)d_in[19];

  const int N = in_sizes[0] / NC;
  const int E = in_sizes[3] / 3;

  float* ws = (float*)d_ws;
  size_t off = 0;
  float* nrm0 = ws + off; off += (size_t)N * NC;
  float* nrm1 = ws + off; off += (size_t)N * NC * 3;
  float* nrm2 = ws + off; off += (size_t)N * NC * 9;
  float* prem = ws + off; off += (size_t)N * 2496;   // premixed W2..W7 node features (~100MB, L2-resident)
  float* comb = ws + off; off += (size_t)N * 2080;
  float* feed = ws + off; off += (size_t)E * 192;
  float* mult = ws + off; off += (size_t)E * 288;

  hipMemsetAsync(comb, 0, (size_t)N * 2080 * sizeof(float), stream);

  norm_kernel<<<N, 128, 0, stream>>>(nf0, nf1, nf2, nrm0, nrm1, nrm2);
  node_premix_kernel<<<N, 32, 0, stream>>>(nrm0, nrm1, nrm2, lin, prem);
  node_mii_kernel<<<N, 32, 0, stream>>>(nrm0, nrm1, nrm2, lin, tpw, comb);
  edge_feed_kernel<<<E, 32, 0, stream>>>(prem, r_ij, eidx, lin, tpw, attw, feed, E);
  dim3 g4((E + 15) / 16, 3);
  attn_wmma_kernel<<<g4, 32, 0, stream>>>(feed, rbf_ij, w1, b1, w2, b2, w3, b3, rbfw, mult, E);
  edge_msg_kernel<<<E, 32, 0, stream>>>(prem, r_ij, eidx, lin, tpw, mult, comb, E);
  node_out_kernel<<<N, 32, 0, stream>>>(comb, mix_w, gate_w, gate_b, (float*)d_out);
}